// SiluJaggedBmmCombineModule_40845138985379
// MI455X (gfx1250) — compile-verified
//
#include <hip/hip_runtime.h>
#include <hip/hip_bf16.h>

typedef __attribute__((ext_vector_type(2))) float v2f;
typedef __attribute__((ext_vector_type(8))) float v8f;

// Problem constants (from reference): N=65536, K=2, E=16, D=512, DOUT=512
constexpr int D      = 512;
constexpr int DOUT   = 512;
constexpr int E      = 16;
constexpr int TILE_M = 64;        // dispatched rows per workgroup
constexpr int BLOCK  = 512;       // 16 waves
constexpr int LDA    = D + 4;     // padded LDS row stride (floats); 516*4B rows, conflict-free

__global__ __launch_bounds__(256)
void moe_zero_out(float4* __restrict__ p, int n4) {
    int i = blockIdx.x * blockDim.x + threadIdx.x;
    if (i < n4) p[i] = make_float4(0.f, 0.f, 0.f, 0.f);
}

// K-loop: 4 row-tiles x 2 col-tiles of 16x16 f32 WMMA per wave.
// MASKED variant zeroes A rows not belonging to expert pass `e` (boundary tiles only).
template <bool MASKED>
__device__ __forceinline__ void gemm_kloop(const float* __restrict__ Wb,
                                           const float* sAp,
                                           int wave, int half, int ml,
                                           const float mrow[4],
                                           v8f acc[4][2])
{
    for (int kk = 0; kk < D; kk += 4) {
        const int krow = kk + half * 2;
        // A fragments: lane(ml,half) -> float2 {A[m][krow], A[m][krow+1]}
        v2f a[4];
        #pragma unroll
        for (int rt = 0; rt < 4; ++rt) {
            a[rt] = *(const v2f*)&sAp[(rt * 16 + ml) * LDA + krow];
            if (MASKED) a[rt] *= mrow[rt];
        }
        // B fragments: {W[krow][c], W[krow+1][c]} (weights are L2-resident)
        const float* Wk = Wb + (size_t)krow * DOUT + ml;
        #pragma unroll
        for (int ct = 0; ct < 2; ++ct) {
            const int c = wave * 16 + ct * 256;
            v2f b;
            b.x = Wk[c];
            b.y = Wk[c + DOUT];
            #pragma unroll
            for (int rt = 0; rt < 4; ++rt)
                acc[rt][ct] = __builtin_amdgcn_wmma_f32_16x16x4_f32(
                    false, a[rt], false, b, (short)0, acc[rt][ct], false, false);
        }
    }
}

__global__ __launch_bounds__(BLOCK)
void moe_grouped_gemm_silu_combine(const int*   __restrict__ offsets,   // [E+1]
                                   const float* __restrict__ jagged,    // [N, D]
                                   const float* __restrict__ weight,    // [E, D, DOUT]
                                   const float* __restrict__ bias,      // [E, DOUT]
                                   const int*   __restrict__ index,     // [T]
                                   const int*   __restrict__ rindex,    // [T]
                                   const float* __restrict__ gates,     // [T]
                                   const int*   __restrict__ gidx,      // [T]
                                   float*       __restrict__ out,       // [N, DOUT]
                                   int T)
{
    __shared__ float sA[TILE_M * LDA];          // ~129 KB of the 320 KB WGP LDS
    __shared__ int   sSrc[TILE_M];
    __shared__ int   sExpert[TILE_M];
    __shared__ int   sDst[TILE_M];
    __shared__ float sGate[TILE_M];

    const int tid  = threadIdx.x;
    const int wave = tid >> 5;        // 0..15
    const int lane = tid & 31;
    const int half = lane >> 4;       // k sub-pair selector (WMMA f32 16x16x4 layout)
    const int ml   = lane & 15;       // row (A) / col (B,C,D) within a 16-tile
    const int r0   = blockIdx.x * TILE_M;

    // ---- per-row metadata (expert id via searchsorted over offsets) ----
    if (tid < TILE_M) {
        int r = r0 + tid;
        if (r < T) {
            sSrc[tid]  = index[r];
            sDst[tid]  = rindex[r];
            sGate[tid] = gates[gidx[r]];
            int e = 0;
            #pragma unroll
            for (int j = 1; j <= E; ++j) e += (r >= offsets[j]) ? 1 : 0;
            sExpert[tid] = e;
        } else {
            sSrc[tid] = 0; sDst[tid] = 0; sGate[tid] = 0.0f; sExpert[tid] = -1;
        }
    }
    __syncthreads();

    // ---- cooperative gather of A tile into LDS (float4, coalesced per row) ----
    {
        constexpr int F4R = D / 4;              // 128 float4 per row
        constexpr int TOT = TILE_M * F4R;       // 8192
        for (int i = tid; i < TOT; i += BLOCK) {
            int row = i / F4R;
            int c4  = i % F4R;
            const float4* src = (const float4*)(jagged + (size_t)sSrc[row] * D);
            float4 v = src[c4];
            *(float4*)&sA[row * LDA + c4 * 4] = v;   // row*2064B is 16B aligned
        }
    }

    // dispatch order is sorted by expert -> experts in this tile are non-decreasing
    const int nvalid = min(TILE_M, T - r0);
    const int e_lo   = sExpert[0];
    const int e_hi   = sExpert[nvalid - 1];
    __syncthreads();

    // ---- accumulators: 4 row-tiles x 2 col-tiles of 16x16 f32 ----
    v8f acc[4][2];
    #pragma unroll
    for (int rt = 0; rt < 4; ++rt)
        #pragma unroll
        for (int ct = 0; ct < 2; ++ct)
            acc[rt][ct] = (v8f){0.f,0.f,0.f,0.f,0.f,0.f,0.f,0.f};

    if (e_lo == e_hi && e_lo >= 0) {
        // fast path (>99% of tiles): single expert, no A masking.
        // padded rows (expert==-1; e_hi comes from the last valid row, never -1) compute garbage that the
        // epilogue skips, so this path is safe even for a partial last tile.
        gemm_kloop<false>(weight + (size_t)e_lo * D * DOUT, sA, wave, half, ml, nullptr, acc);
    } else {
        // boundary tile: one masked pass per expert; zeroed A rows contribute nothing,
        // so all passes accumulate into the same C.
        const int elo = (e_lo < 0) ? 0 : e_lo;
        for (int e = elo; e <= e_hi; ++e) {
            float mrow[4];
            #pragma unroll
            for (int rt = 0; rt < 4; ++rt)
                mrow[rt] = (sExpert[rt * 16 + ml] == e) ? 1.0f : 0.0f;
            gemm_kloop<true>(weight + (size_t)e * D * DOUT, sA, wave, half, ml, mrow, acc);
        }
    }

    // ---- epilogue: bias + SiLU + gate + scatter-add combine ----
    #pragma unroll
    for (int rt = 0; rt < 4; ++rt) {
        #pragma unroll
        for (int ct = 0; ct < 2; ++ct) {
            const int n = wave * 16 + ct * 256 + ml;
            #pragma unroll
            for (int j = 0; j < 8; ++j) {
                const int m  = rt * 16 + half * 8 + j;   // row within tile
                const int em = sExpert[m];
                if (em < 0) continue;
                float h = acc[rt][ct][j] + bias[em * DOUT + n];
                h = h / (1.0f + __expf(-h));             // SiLU
                h *= sGate[m];
                atomicAdd(out + (size_t)sDst[m] * DOUT + n, h);
            }
        }
    }
}

extern "C" void kernel_launch(void* const* d_in, const int* in_sizes, int n_in,
                              void* d_out, int out_size, void* d_ws, size_t ws_size,
                              hipStream_t stream) {
    const int*   offsets = (const int*)  d_in[0];
    const float* jagged  = (const float*)d_in[1];
    const float* weight  = (const float*)d_in[2];
    const float* bias    = (const float*)d_in[3];
    const int*   index   = (const int*)  d_in[4];
    const int*   rindex  = (const int*)  d_in[5];
    const float* gates   = (const float*)d_in[6];
    const int*   gidx    = (const int*)  d_in[7];
    float*       out     = (float*)d_out;

    const int T = in_sizes[4];

    // zero-init output (combine is a scatter-add)
    const int n4 = out_size / 4;
    moe_zero_out<<<(n4 + 255) / 256, 256, 0, stream>>>((float4*)out, n4);

    const int nTiles = (T + TILE_M - 1) / TILE_M;
    moe_grouped_gemm_silu_combine<<<nTiles, BLOCK, 0, stream>>>(
        offsets, jagged, weight, bias, index, rindex, gates, gidx, out, T);
}